// HGCNLayer_90366111908553
// MI455X (gfx1250) — compile-verified
//
#include <hip/hip_runtime.h>
#include <cstdint>

// ---------------------------------------------------------------------------
// HGCN layer (KG segment-mean + COO spmm segment-sum), MI455X / gfx1250.
// Memory-bound scatter/gather: wave32 row mapping (lane = float4 of a 128-f32
// row), b128 loads, f32 global atomics, async global->LDS staging of the
// 32x128 relation-weight table.
// ---------------------------------------------------------------------------

typedef float v4f __attribute__((ext_vector_type(4)));
typedef int   v4i __attribute__((vector_size(16)));   // matches builtin param

#define D     128
#define DV    32          // D / 4 floats per lane-vector
#define NREL  32          // relation table rows (16 KB total)

#define GLOBAL_AS __attribute__((address_space(1)))
#define LDS_AS    __attribute__((address_space(3)))

typedef GLOBAL_AS v4i* gptr_v4i;   // global (addrspace 1) int4*
typedef LDS_AS    v4i* lptr_v4i;   // LDS    (addrspace 3) int4*

#if defined(__gfx1250__) && __has_builtin(__builtin_amdgcn_global_load_async_to_lds_b128)
#define USE_ASYNC_LDS 1
#else
#define USE_ASYNC_LDS 0
#endif

__device__ __forceinline__ void atomic_add_f32(float* p, float v) {
  // agent-scope relaxed fadd -> global_atomic_add_f32 (no CAS expansion)
  __hip_atomic_fetch_add(p, v, __ATOMIC_RELAXED, __HIP_MEMORY_SCOPE_AGENT);
}

// ---------------------------------------------------------------------------
// Zero-fill (float4 granularity). Harness poisons d_out / d_ws with 0xAA.
// ---------------------------------------------------------------------------
__global__ void zero_f32x4(float* __restrict__ p, int n4) {
  int i = blockIdx.x * blockDim.x + threadIdx.x;
  if (i < n4) {
    v4f z = {0.0f, 0.0f, 0.0f, 0.0f};
    ((v4f*)p)[i] = z;
  }
}

// ---------------------------------------------------------------------------
// KG edge kernel: wave per edge, grid-stride. Relation table staged in LDS
// via CDNA5 async global->LDS copies.
//   out_entity[head] += emb[tail] * weight[etype-1];  counts[head] += 1
// ---------------------------------------------------------------------------
__global__ void __launch_bounds__(256)
kg_edge_kernel(const float* __restrict__ emb,
               const int*   __restrict__ eidx,    // [2*E] row0=head, row1=tail
               const int*   __restrict__ etype,   // [E], values 1..NREL
               const float* __restrict__ weight,  // [NREL*D]
               float*       __restrict__ out_entity,
               float*       __restrict__ counts,
               int n_edges)
{
  __shared__ v4f wsh[NREL * DV];                  // 16 KB
  const v4f* w4 = (const v4f*)weight;

  for (int i = threadIdx.x; i < NREL * DV; i += blockDim.x) {
#if USE_ASYNC_LDS
    __builtin_amdgcn_global_load_async_to_lds_b128(
        (gptr_v4i)(uintptr_t)(w4 + i),
        (lptr_v4i)(uint32_t)(uintptr_t)(&wsh[i]),
        0, 0);
#else
    wsh[i] = w4[i];
#endif
  }
#if USE_ASYNC_LDS
#if __has_builtin(__builtin_amdgcn_s_wait_asynccnt)
  __builtin_amdgcn_s_wait_asynccnt(0);
#else
  asm volatile("s_wait_asynccnt 0" ::: "memory");
#endif
#endif
  __syncthreads();

  const int lane   = threadIdx.x & 31;
  const int wave   = (blockIdx.x * blockDim.x + threadIdx.x) >> 5;
  const int nwaves = (gridDim.x * blockDim.x) >> 5;
  const v4f* emb4  = (const v4f*)emb;

  for (int e = wave; e < n_edges; e += nwaves) {
    int head = eidx[e];
    int tail = eidx[n_edges + e];
    int t    = etype[e] - 1;

    v4f v = emb4[(size_t)tail * DV + lane] * wsh[t * DV + lane];

    float* dst = out_entity + (size_t)head * D + lane * 4;
    atomic_add_f32(dst + 0, v.x);
    atomic_add_f32(dst + 1, v.y);
    atomic_add_f32(dst + 2, v.z);
    atomic_add_f32(dst + 3, v.w);
    if (lane == 0) atomic_add_f32(&counts[head], 1.0f);
  }
}

// ---------------------------------------------------------------------------
// User aggregate: wave per nnz.  out_user[row] += val * emb[col]
// ---------------------------------------------------------------------------
__global__ void __launch_bounds__(256)
user_agg_kernel(const float* __restrict__ emb,
                const int*   __restrict__ rows,
                const int*   __restrict__ cols,
                const float* __restrict__ vals,
                float*       __restrict__ out_user,
                int nnz)
{
  const int lane = threadIdx.x & 31;
  const int e    = (blockIdx.x * blockDim.x + threadIdx.x) >> 5;
  if (e >= nnz) return;

  int   r = rows[e];
  int   c = cols[e];
  float s = vals[e];

  v4f v = ((const v4f*)emb)[(size_t)c * DV + lane] * s;

  float* dst = out_user + (size_t)r * D + lane * 4;
  atomic_add_f32(dst + 0, v.x);
  atomic_add_f32(dst + 1, v.y);
  atomic_add_f32(dst + 2, v.z);
  atomic_add_f32(dst + 3, v.w);
}

// ---------------------------------------------------------------------------
// Mean normalize: wave per entity row; out /= max(count, 1)
// ---------------------------------------------------------------------------
__global__ void __launch_bounds__(256)
normalize_kernel(float* __restrict__ out_entity,
                 const float* __restrict__ counts,
                 int n_entities)
{
  const int lane = threadIdx.x & 31;
  const int row  = (blockIdx.x * blockDim.x + threadIdx.x) >> 5;
  if (row >= n_entities) return;

  float c   = counts[row];
  float inv = 1.0f / fmaxf(c, 1.0f);

  v4f* p = (v4f*)out_entity + (size_t)row * DV + lane;
  *p = *p * inv;
}

// ---------------------------------------------------------------------------
// Launch
// ---------------------------------------------------------------------------
extern "C" void kernel_launch(void* const* d_in, const int* in_sizes, int n_in,
                              void* d_out, int out_size, void* d_ws, size_t ws_size,
                              hipStream_t stream)
{
  const float* emb    = (const float*)d_in[0];
  const int*   eidx   = (const int*)  d_in[1];
  const int*   etype  = (const int*)  d_in[2];
  const int*   irows  = (const int*)  d_in[3];
  const int*   icols  = (const int*)  d_in[4];
  const float* ivals  = (const float*)d_in[5];
  // d_in[6] = n_users scalar (device); derived on host instead (capture-safe)
  const float* weight = (const float*)d_in[7];

  const int n_entities = in_sizes[0] / D;   // 100000
  const int n_edges    = in_sizes[2];       // 600000
  const int nnz        = in_sizes[5];       // 1000000

  float* out_entity = (float*)d_out;
  float* out_user   = out_entity + (size_t)n_entities * D;
  float* counts     = (float*)d_ws;         // n_entities floats of scratch

  // 1) zero output (entity_agg ++ user_agg) and the counts scratch
  {
    int n4 = out_size / 4;
    zero_f32x4<<<(n4 + 255) / 256, 256, 0, stream>>>((float*)d_out, n4);
    int c4 = (n_entities + 3) / 4;
    zero_f32x4<<<(c4 + 255) / 256, 256, 0, stream>>>(counts, c4);
  }

  // 2) KG edge scatter (grid-stride; 2048 blocks => staging cost 32 MB total)
  kg_edge_kernel<<<2048, 256, 0, stream>>>(emb, eidx, etype, weight,
                                           out_entity, counts, n_edges);

  // 3) user COO scatter (wave per nnz)
  {
    long long waves  = (long long)nnz;
    int       blocks = (int)((waves * 32 + 255) / 256);
    user_agg_kernel<<<blocks, 256, 0, stream>>>(emb, irows, icols, ivals,
                                                out_user, nnz);
  }

  // 4) mean normalize (stream order guarantees scatter completion)
  {
    long long waves  = (long long)n_entities;
    int       blocks = (int)((waves * 32 + 255) / 256);
    normalize_kernel<<<blocks, 256, 0, stream>>>(out_entity, counts, n_entities);
  }
}